// SwinTransformerBlock_33578054320258
// MI455X (gfx1250) — compile-verified
//
#include <hip/hip_runtime.h>

// ---------------- problem constants (fixed by the reference) ----------------
#define WSZ      7
#define SHIFT    3
#define NHEADS   4
#define CDIM     128
#define HDIM     32          // head dim
#define NTOK     49          // tokens per window (7*7)
#define HW       56
#define NWIN_IMG 64          // (56/7)^2
#define BATCH    32
#define NWINDOWS (BATCH * NWIN_IMG)   // 2048

typedef float v2f __attribute__((ext_vector_type(2)));
typedef float v8f __attribute__((ext_vector_type(8)));

// ---------------- LDS layout (floats), strides padded to kill bank conflicts ----
#define QKVS_STRIDE 388                       // 384 + 4  (388 % 64 == 4)
#define XOS_STRIDE  132                       // 128 + 4
#define ATTN_STRIDE 68                        // 64 + 4
#define QKVS_OFF    0
#define XOS_OFF     (NTOK * QKVS_STRIDE)
#define ATTN_OFF    (XOS_OFF + NTOK * XOS_STRIDE)
#define ATTN_HSTR   (NTOK * ATTN_STRIDE)
#define LDS_FLOATS  (ATTN_OFF + NHEADS * ATTN_HSTR)   // 38808 floats -> 155232 B

// fp32 WMMA: D = A(16x4) * B(4x16) + C(16x16), exact fp32 (matches reference)
__device__ __forceinline__ v8f wmma4(v2f a, v2f b, v8f c) {
  return __builtin_amdgcn_wmma_f32_16x16x4_f32(
      /*neg_a=*/false, a, /*neg_b=*/false, b,
      /*c_mod=*/(short)0, c, /*reuse_a=*/false, /*reuse_b=*/false);
}

__device__ __forceinline__ void zero_acc(v8f* acc) {
#pragma unroll
  for (int i = 0; i < 4; ++i)
#pragma unroll
    for (int j = 0; j < 8; ++j) acc[i][j] = 0.0f;
}

__global__ void __launch_bounds__(256)
swin_block_fused(const float* __restrict__ x,
                 const float* __restrict__ attn_mask,   // (64, 49, 49)
                 const float* __restrict__ qkv_w,       // (384, 128)
                 const float* __restrict__ qkv_b,       // (384)
                 const float* __restrict__ proj_w,      // (128, 128)
                 const float* __restrict__ proj_b,      // (128)
                 const float* __restrict__ rel_bias,    // (169, 4)
                 float* __restrict__ out)               // (32, 3136, 128)
{
  extern __shared__ float smem[];
  float* qkvs = smem;              // [49][388] : q | k | v per row
  float* xos  = smem + XOS_OFF;    // [49][132] : X during QKV, O after attn
  float* attn = smem + ATTN_OFF;   // [4][49][68]

  const int w    = blockIdx.x;
  const int b    = w >> 6;         // image
  const int wimg = w & 63;         // window within image
  const int wy   = wimg >> 3, wx = wimg & 7;
  const int tid  = threadIdx.x;
  const int lane = tid & 31;
  const int wv   = tid >> 5;       // 8 waves
  const float scale = 0.17677669529663687f;  // 32^-0.5

  const int lrow   = lane & 15;          // M/N index inside a 16-wide tile
  const int khalf  = (lane >> 4) << 1;   // K sub-offset {0,2} per lane half
  const int mbase8 = (lane >> 4) << 3;   // C/D: M offset {0,8} per lane half

  // clamped A-matrix rows per M-tile (loop-invariant; garbage rows are
  // dropped at store time, only the address must stay in-bounds)
  int mrowc[4];
#pragma unroll
  for (int mt = 0; mt < 4; ++mt) {
    const int m = mt * 16 + lrow;
    mrowc[mt] = (m < NTOK) ? m : 0;
  }

  // ---------------- 1) load shifted window -> xos ----------------
  for (int i = tid; i < NTOK * (CDIM / 4); i += 256) {
    int t  = i >> 5;                 // token 0..48
    int c4 = i & 31;                 // float4 index 0..31
    int r = t / WSZ, cc = t % WSZ;
    int sh = (wy * WSZ + r + SHIFT) % HW;
    int sw = (wx * WSZ + cc + SHIFT) % HW;
    const float4 v = ((const float4*)(x + ((size_t)b * HW * HW + sh * HW + sw) * CDIM))[c4];
    float* dst = xos + t * XOS_STRIDE + c4 * 4;
    dst[0] = v.x; dst[1] = v.y; dst[2] = v.z; dst[3] = v.w;
  }
  __syncthreads();

  // ---------------- 2) QKV GEMM: (49x128) @ (128x384^T) ----------------
  // wave wv owns n-tiles {3wv .. 3wv+2}; 4 M-tiles each; K = 128 -> 32 steps.
  for (int nti = 0; nti < 3; ++nti) {
    const int ncol = (wv * 3 + nti) * 16 + lrow;
    const float* __restrict__ brow = qkv_w + (size_t)ncol * CDIM;
    v8f acc[4]; zero_acc(acc);
#pragma unroll 4
    for (int ks = 0; ks < 32; ++ks) {
      const int k0 = (ks << 2) + khalf;
      const v2f bfrag = *(const v2f*)(brow + k0);
#pragma unroll
      for (int mt = 0; mt < 4; ++mt) {
        const v2f afrag = *(const v2f*)(xos + mrowc[mt] * XOS_STRIDE + k0);
        acc[mt] = wmma4(afrag, bfrag, acc[mt]);
      }
    }
    const float bn  = qkv_b[ncol];
    const float mul = (ncol < CDIM) ? scale : 1.0f;   // q gets 1/sqrt(hd)
#pragma unroll
    for (int mt = 0; mt < 4; ++mt)
#pragma unroll
      for (int j = 0; j < 8; ++j) {
        const int m = mt * 16 + mbase8 + j;
        if (m < NTOK) qkvs[m * QKVS_STRIDE + ncol] = (acc[mt][j] + bn) * mul;
      }
  }
  __syncthreads();

  // ---------------- 3) S = q @ k^T  (+bias +mask) ----------------
  const int h    = wv >> 1;        // head, 2 waves per head
  const int half = wv & 1;
  const float* qh = qkvs + h * HDIM;
  const float* kh = qkvs + CDIM + h * HDIM;
  const float* vh = qkvs + 2 * CDIM + h * HDIM;
  float* attnh = attn + h * ATTN_HSTR;

  for (int nti = 0; nti < 2; ++nti) {
    const int n  = (half * 2 + nti) * 16 + lrow;     // key token, 0..63
    const int ncl = (n < NTOK) ? n : 0;              // clamped (garbage cols -> 0 later)
    const float* __restrict__ brow = kh + ncl * QKVS_STRIDE;
    v8f acc[4]; zero_acc(acc);
#pragma unroll
    for (int ks = 0; ks < 8; ++ks) {                 // K = 32
      const int k0 = (ks << 2) + khalf;
      const v2f bfrag = *(const v2f*)(brow + k0);
#pragma unroll
      for (int mt = 0; mt < 4; ++mt) {
        const v2f afrag = *(const v2f*)(qh + mrowc[mt] * QKVS_STRIDE + k0);
        acc[mt] = wmma4(afrag, bfrag, acc[mt]);
      }
    }
    // epilogue: + relative-position bias + shift mask; zero-fill n>=49
    const int nr = ncl / WSZ, nc = ncl % WSZ;
    const bool nvalid = (n < NTOK);
#pragma unroll
    for (int mt = 0; mt < 4; ++mt)
#pragma unroll
      for (int j = 0; j < 8; ++j) {
        const int m = mt * 16 + mbase8 + j;
        if (m >= NTOK) continue;                     // rows 49..63 not stored
        const int mr = m / WSZ, mc = m % WSZ;
        const int rpi = (mr - nr + WSZ - 1) * (2 * WSZ - 1) + (mc - nc + WSZ - 1);
        const float add = rel_bias[rpi * NHEADS + h]
                        + attn_mask[((size_t)wimg * NTOK + m) * NTOK + ncl];
        attnh[m * ATTN_STRIDE + n] = nvalid ? (acc[mt][j] + add) : 0.0f;
      }
  }
  __syncthreads();

  // ---------------- 4) softmax, one thread per (head,row) ----------------
  if (tid < NHEADS * NTOK) {
    const int hh = tid / NTOK, m = tid % NTOK;
    float* row = attn + hh * ATTN_HSTR + m * ATTN_STRIDE;
    float mx = row[0];
    for (int i = 1; i < NTOK; ++i) mx = fmaxf(mx, row[i]);
    float s = 0.0f;
    for (int i = 0; i < NTOK; ++i) { float e = __expf(row[i] - mx); row[i] = e; s += e; }
    const float inv = 1.0f / s;
    for (int i = 0; i < NTOK; ++i) row[i] *= inv;
  }
  __syncthreads();

  // ---------------- 5) O = P @ V  (K = 49 zero-padded to 64) ----------------
  {
    const int n = half * 16 + lrow;                  // 0..31 within head dim
    v8f acc[4]; zero_acc(acc);
    // k-steps 0..11: all kk < 49, unguarded
#pragma unroll
    for (int ks = 0; ks < 12; ++ks) {
      const int k0 = (ks << 2) + khalf;
      v2f bfrag;
      bfrag[0] = vh[(k0)     * QKVS_STRIDE + n];
      bfrag[1] = vh[(k0 + 1) * QKVS_STRIDE + n];
#pragma unroll
      for (int mt = 0; mt < 4; ++mt) {
        const v2f afrag = *(const v2f*)(attnh + mrowc[mt] * ATTN_STRIDE + k0);
        acc[mt] = wmma4(afrag, bfrag, acc[mt]);
      }
    }
    // k-steps 12..15: V rows kk >= 49 must contribute true zeros (K padding)
#pragma unroll
    for (int ks = 12; ks < 16; ++ks) {
      const int k0  = (ks << 2) + khalf;
      const int kc0 = (k0     < NTOK) ? k0     : 0;
      const int kc1 = (k0 + 1 < NTOK) ? k0 + 1 : 0;
      const float t0 = vh[kc0 * QKVS_STRIDE + n];
      const float t1 = vh[kc1 * QKVS_STRIDE + n];
      v2f bfrag;
      bfrag[0] = (k0     < NTOK) ? t0 : 0.0f;
      bfrag[1] = (k0 + 1 < NTOK) ? t1 : 0.0f;
#pragma unroll
      for (int mt = 0; mt < 4; ++mt) {
        const v2f afrag = *(const v2f*)(attnh + mrowc[mt] * ATTN_STRIDE + k0);
        acc[mt] = wmma4(afrag, bfrag, acc[mt]);       // attn cols 49..63 are real zeros
      }
    }
#pragma unroll
    for (int mt = 0; mt < 4; ++mt)
#pragma unroll
      for (int j = 0; j < 8; ++j) {
        const int m = mt * 16 + mbase8 + j;
        if (m < NTOK) xos[m * XOS_STRIDE + h * HDIM + n] = acc[mt][j];
      }
  }
  __syncthreads();

  // ---------------- 6) projection + window-reverse + un-roll store ----------
  {
    const int ncol = wv * 16 + lrow;                 // 8 n-tiles across 8 waves
    const float* __restrict__ brow = proj_w + (size_t)ncol * CDIM;
    v8f acc[4]; zero_acc(acc);
#pragma unroll 4
    for (int ks = 0; ks < 32; ++ks) {
      const int k0 = (ks << 2) + khalf;
      const v2f bfrag = *(const v2f*)(brow + k0);
#pragma unroll
      for (int mt = 0; mt < 4; ++mt) {
        const v2f afrag = *(const v2f*)(xos + mrowc[mt] * XOS_STRIDE + k0);
        acc[mt] = wmma4(afrag, bfrag, acc[mt]);
      }
    }
    const float pb = proj_b[ncol];
#pragma unroll
    for (int mt = 0; mt < 4; ++mt)
#pragma unroll
      for (int j = 0; j < 8; ++j) {
        const int m = mt * 16 + mbase8 + j;
        if (m >= NTOK) continue;
        const int r = m / WSZ, cc = m % WSZ;
        // reverse shift cancels forward shift: dest == source coordinates
        const int sh = (wy * WSZ + r + SHIFT) % HW;
        const int sw = (wx * WSZ + cc + SHIFT) % HW;
        out[((size_t)b * HW * HW + sh * HW + sw) * CDIM + ncol] = acc[mt][j] + pb;
      }
  }
}

extern "C" void kernel_launch(void* const* d_in, const int* in_sizes, int n_in,
                              void* d_out, int out_size, void* d_ws, size_t ws_size,
                              hipStream_t stream) {
  const float* x    = (const float*)d_in[0];
  const float* mask = (const float*)d_in[1];
  const float* qw   = (const float*)d_in[2];
  const float* qb   = (const float*)d_in[3];
  const float* pw   = (const float*)d_in[4];
  const float* pb   = (const float*)d_in[5];
  const float* rbt  = (const float*)d_in[6];
  float* out = (float*)d_out;
  (void)in_sizes; (void)n_in; (void)out_size; (void)d_ws; (void)ws_size;

  const size_t lds_bytes = (size_t)LDS_FLOATS * sizeof(float);  // ~152 KB -> 2 blocks/WGP
  (void)hipFuncSetAttribute((const void*)swin_block_fused,
                            hipFuncAttributeMaxDynamicSharedMemorySize,
                            (int)lds_bytes);
  swin_block_fused<<<NWINDOWS, 256, lds_bytes, stream>>>(x, mask, qw, qb, pw, pb, rbt, out);
}